// OldWhileLoopChoiProjection_54941221651083
// MI455X (gfx1250) — compile-verified
//
#include <hip/hip_runtime.h>
#include <hip/hip_bf16.h>

#define DD   4
#define D2   16
#define PIT  17          // LDS row pitch (floats) to avoid bank conflicts
#define NSWEEP 9
#define TOLF 1e-8f

typedef float v2f __attribute__((ext_vector_type(2)));
typedef float v8f __attribute__((ext_vector_type(8)));

__device__ __forceinline__ v8f wmma4(v2f a, v2f b, v8f c) {
  // V_WMMA_F32_16X16X4_F32 : D = A(16x4) * B(4x16) + C(16x16)
  return __builtin_amdgcn_wmma_f32_16x16x4_f32(false, a, false, b, (short)0, c,
                                               false, false);
}

// ---- fragment loads per CDNA5 ISA VGPR layouts (wave32) ----
// A (16x4): lanes 0-15 -> K {0,1}, lanes 16-31 -> K {2,3}; row = lane&15
__device__ __forceinline__ v2f frag_a(const float* m, int kb, int lane) {
  int r = lane & 15, kh = (lane >> 4) << 1;
  int k0 = kb * 4 + kh;
  v2f f; f.x = m[r * PIT + k0]; f.y = m[r * PIT + k0 + 1]; return f;
}
__device__ __forceinline__ v2f frag_a_scal(const float* m, const float* s,
                                           int kb, int lane) {
  int r = lane & 15, kh = (lane >> 4) << 1;
  int k0 = kb * 4 + kh;
  v2f f; f.x = m[r * PIT + k0] * s[k0]; f.y = m[r * PIT + k0 + 1] * s[k0 + 1];
  return f;
}
// B (4x16), row-striped across lanes: VGPR0 -> K0/K2, VGPR1 -> K1/K3
__device__ __forceinline__ v2f frag_b(const float* m, int kb, int lane) {
  int n = lane & 15, kh = (lane >> 4) << 1;
  int k0 = kb * 4 + kh;
  v2f f; f.x = m[k0 * PIT + n]; f.y = m[(k0 + 1) * PIT + n]; return f;
}
// B = srcᵀ   (B[k][n] = src[n][k]) — used for X·Vᴴ style operands
__device__ __forceinline__ v2f frag_bt(const float* m, int kb, int lane) {
  int n = lane & 15, kh = (lane >> 4) << 1;
  int k0 = kb * 4 + kh;
  v2f f; f.x = m[n * PIT + k0]; f.y = m[n * PIT + k0 + 1]; return f;
}
// C/D (16x16): VGPR r -> row r (lanes 0-15) / row r+8 (lanes 16-31)
__device__ __forceinline__ void store_c(float* m, v8f c, int lane) {
  int n = lane & 15, mb = (lane >> 4) * 8;
#pragma unroll
  for (int r = 0; r < 8; ++r) m[(mb + r) * PIT + n] = c[r];
}

// complex C = A * B (all row-major in LDS)
__device__ __forceinline__ void cmm_nn(const float* ar, const float* ai,
                                       const float* br, const float* bi,
                                       float* cr, float* ci, int lane) {
  v8f accr = {}; v8f acci = {};
#pragma unroll
  for (int kb = 0; kb < 4; ++kb) {
    v2f af = frag_a(ar, kb, lane), aif = frag_a(ai, kb, lane);
    v2f bf = frag_b(br, kb, lane), bif = frag_b(bi, kb, lane);
    accr = wmma4(af, bf, accr);
    v2f nai; nai.x = -aif.x; nai.y = -aif.y;
    accr = wmma4(nai, bif, accr);         // Cr = Ar*Br - Ai*Bi
    acci = wmma4(af, bif, acci);
    acci = wmma4(aif, bf, acci);          // Ci = Ar*Bi + Ai*Br
  }
  store_c(cr, accr, lane);
  store_c(ci, acci, lane);
}

__global__ __launch_bounds__(32)
void choi_dykstra_kernel(const float* __restrict__ xre,
                         const float* __restrict__ xim,
                         const int* __restrict__ maxit_p,
                         float* __restrict__ out) {
  __shared__ float Xr[D2 * PIT], Xi[D2 * PIT];
  __shared__ float Pr[D2 * PIT], Pi[D2 * PIT];
  __shared__ float Qr[D2 * PIT], Qi[D2 * PIT];
  __shared__ float Ar[D2 * PIT], Ai[D2 * PIT];   // eigh work
  __shared__ float Vr[D2 * PIT], Vi[D2 * PIT];
  __shared__ float Wr[D2 * PIT], Wi[D2 * PIT];   // T / Z / Mbig scratch
  __shared__ float ev[16], sortv[16], lam[16];
  __shared__ float rc[8], rsr[8], rsi[8];
  __shared__ float m4r[16], m4i[16];
  __shared__ float a4r[16], a4i[16], u4r[16], u4i[16], w4[4];
  __shared__ float scal[2];

  const int lane = threadIdx.x;
  const long b = blockIdx.x;
  const float* inr = xre + b * 256;
  const float* ini = xim + b * 256;

  // stage raw input in LDS
#pragma unroll
  for (int t = 0; t < 8; ++t) {
    int e = lane + 32 * t; int i = e >> 4, j = e & 15;
    Wr[i * PIT + j] = inr[e];
    Wi[i * PIT + j] = ini[e];
  }
  __syncthreads();
  // Hermitize: X = 0.5 (A + A^H); p = q = 0
#pragma unroll
  for (int t = 0; t < 8; ++t) {
    int e = lane + 32 * t; int i = e >> 4, j = e & 15; int o = i * PIT + j;
    Xr[o] = 0.5f * (Wr[o] + Wr[j * PIT + i]);
    Xi[o] = 0.5f * (Wi[o] - Wi[j * PIT + i]);
    Pr[o] = 0.f; Pi[o] = 0.f; Qr[o] = 0.f; Qi[o] = 0.f;
  }
  __syncthreads();

  const int iters = maxit_p[0] + 1;   // while (k <= max_iter)
  for (int it = 0; it < iters; ++it) {
    float dsum = 0.f;
    // ---- T = X + p  -> W
    for (int t = 0; t < 8; ++t) {
      int e = lane + 32 * t; int i = e >> 4, j = e & 15; int o = i * PIT + j;
      Wr[o] = Xr[o] + Pr[o];
      Wi[o] = Xi[o] + Pi[o];
    }
    __syncthreads();
    // ---- M4 = I/4 - partial_trace(T)
    if (lane < 16) {
      int j = lane >> 2, k = lane & 3;
      float sr = 0.f, si = 0.f;
#pragma unroll
      for (int a = 0; a < 4; ++a) {
        sr += Wr[(4 * a + j) * PIT + 4 * a + k];
        si += Wi[(4 * a + j) * PIT + 4 * a + k];
      }
      m4r[lane] = (j == k ? 0.25f : 0.f) - sr;
      m4i[lane] = -si;
    }
    __syncthreads();
    // ---- Y = T + corr/4; p_new = -corr/4; Z = Y + q -> W and A; V = I
    for (int t = 0; t < 8; ++t) {
      int e = lane + 32 * t; int i = e >> 4, j = e & 15; int o = i * PIT + j;
      float cr = 0.f, cim = 0.f;
      if ((i >> 2) == (j >> 2)) {
        int idx = (i & 3) * 4 + (j & 3);
        cr = 0.25f * m4r[idx]; cim = 0.25f * m4i[idx];
      }
      float yr = Wr[o] + cr, yi = Wi[o] + cim;
      float npr = -cr, npi = -cim;
      float dr = npr - Pr[o], di = npi - Pi[o];
      dsum += dr * dr + di * di;
      Pr[o] = npr; Pi[o] = npi;
      float zr = yr + Qr[o], zi = yi + Qi[o];
      Wr[o] = zr; Wi[o] = zi;     // keep Z for q update
      Ar[o] = zr; Ai[o] = zi;     // eigh input
      Vr[o] = (i == j) ? 1.f : 0.f; Vi[o] = 0.f;
    }
    __syncthreads();

    // ---- complex Hermitian Jacobi eigh, parallel (tournament) ordering
    for (int sw = 0; sw < NSWEEP; ++sw) {
      for (int rnd = 0; rnd < 15; ++rnd) {
        if (lane < 8) {
          int p, q;
          if (lane == 0) { p = 15; q = rnd; }
          else { p = (rnd + lane) % 15; q = (rnd + 15 - lane) % 15; }
          if (p > q) { int tm = p; p = q; q = tm; }
          float app = Ar[p * PIT + p], aqq = Ar[q * PIT + q];
          float br_ = Ar[p * PIT + q], bi_ = Ai[p * PIT + q];
          float h = sqrtf(br_ * br_ + bi_ * bi_);
          float c = 1.f, sr = 0.f, si = 0.f;
          if (h > 1e-20f) {
            float tau = (aqq - app) / (2.f * h);
            float tt = (tau >= 0.f ? 1.f : -1.f) /
                       (fabsf(tau) + sqrtf(1.f + tau * tau));
            c = rsqrtf(1.f + tt * tt);
            float sg = tt * c;
            sr = sg * br_ / h; si = sg * bi_ / h;
          }
          rc[lane] = c; rsr[lane] = sr; rsi[lane] = si;
        }
        __syncthreads();
        // column phase: B = A*G and V = V*G  (columns disjoint across pairs)
#pragma unroll
        for (int t = 0; t < 4; ++t) {
          int id = lane + 32 * t; int pi = id >> 4, r = id & 15;
          int p, q;
          if (pi == 0) { p = 15; q = rnd; }
          else { p = (rnd + pi) % 15; q = (rnd + 15 - pi) % 15; }
          if (p > q) { int tm = p; p = q; q = tm; }
          float c = rc[pi], sr = rsr[pi], si = rsi[pi];
          float xpr = Ar[r * PIT + p], xpi = Ai[r * PIT + p];
          float xqr = Ar[r * PIT + q], xqi = Ai[r * PIT + q];
          Ar[r * PIT + p] = c * xpr + (sr * xqr + si * xqi);
          Ai[r * PIT + p] = c * xpi + (sr * xqi - si * xqr);
          Ar[r * PIT + q] = c * xqr - (sr * xpr - si * xpi);
          Ai[r * PIT + q] = c * xqi - (sr * xpi + si * xpr);
          float vpr = Vr[r * PIT + p], vpi = Vi[r * PIT + p];
          float vqr = Vr[r * PIT + q], vqi = Vi[r * PIT + q];
          Vr[r * PIT + p] = c * vpr + (sr * vqr + si * vqi);
          Vi[r * PIT + p] = c * vpi + (sr * vqi - si * vqr);
          Vr[r * PIT + q] = c * vqr - (sr * vpr - si * vpi);
          Vi[r * PIT + q] = c * vqi - (sr * vpi + si * vpr);
        }
        __syncthreads();
        // row phase: A' = G^H * B  (rows disjoint across pairs)
#pragma unroll
        for (int t = 0; t < 4; ++t) {
          int id = lane + 32 * t; int pi = id >> 4, cl = id & 15;
          int p, q;
          if (pi == 0) { p = 15; q = rnd; }
          else { p = (rnd + pi) % 15; q = (rnd + 15 - pi) % 15; }
          if (p > q) { int tm = p; p = q; q = tm; }
          float c = rc[pi], sr = rsr[pi], si = rsi[pi];
          float ypr = Ar[p * PIT + cl], ypi = Ai[p * PIT + cl];
          float yqr = Ar[q * PIT + cl], yqi = Ai[q * PIT + cl];
          Ar[p * PIT + cl] = c * ypr + (sr * yqr - si * yqi);
          Ai[p * PIT + cl] = c * ypi + (sr * yqi + si * yqr);
          Ar[q * PIT + cl] = c * yqr - (sr * ypr + si * ypi);
          Ai[q * PIT + cl] = c * yqi - (sr * ypi - si * ypr);
        }
        __syncthreads();
      }
    }

    // ---- eigenvalue water-filling (descending sort + shift)
    if (lane < 16) ev[lane] = Ar[lane * PIT + lane];
    __syncthreads();
    if (lane < 16) {
      float v = ev[lane];
      int rank = 0;
      for (int j2 = 0; j2 < 16; ++j2) {
        float u = ev[j2];
        if (u > v || (u == v && j2 < lane)) ++rank;
      }
      sortv[rank] = v;
    }
    __syncthreads();
    if (lane == 0) {
      float total = 0.f;
      for (int i2 = 0; i2 < 16; ++i2) total += sortv[i2];
      int ifin = -1; float tail_at = 0.f; float run = 0.f;
      for (int i2 = 0; i2 < 16; ++i2) {
        run += sortv[i2];
        float tail = total - run;
        if (sortv[i2] + tail / (float)(i2 + 1) >= 0.f) { ifin = i2; tail_at = tail; }
      }
      scal[0] = (ifin >= 0) ? tail_at / (float)(ifin + 1) : 0.f;
      scal[1] = (ifin >= 0) ? sortv[ifin] : 3.4e38f;
    }
    __syncthreads();
    if (lane < 16) {
      float v = ev[lane];
      lam[lane] = (v >= scal[1]) ? (v + scal[0]) : 0.f;
    }
    __syncthreads();

    // ---- Xn = (V * diag(lam)) * V^H  via WMMA (lam folded into A-frag load)
    {
      v8f accr = {}; v8f acci = {};
#pragma unroll
      for (int kb = 0; kb < 4; ++kb) {
        v2f wr = frag_a_scal(Vr, lam, kb, lane);
        v2f wi = frag_a_scal(Vi, lam, kb, lane);
        v2f btr = frag_bt(Vr, kb, lane);     // (V^H)r = Vrᵀ
        v2f bti = frag_bt(Vi, kb, lane);     // (V^H)i = -Viᵀ
        accr = wmma4(wr, btr, accr);
        accr = wmma4(wi, bti, accr);         // Cr = WrVrᵀ + WiViᵀ
        acci = wmma4(wi, btr, acci);
        v2f nwr; nwr.x = -wr.x; nwr.y = -wr.y;
        acci = wmma4(nwr, bti, acci);        // Ci = WiVrᵀ - WrViᵀ
      }
      store_c(Xr, accr, lane);
      store_c(Xi, acci, lane);
    }
    __syncthreads();
    // ---- q_new = Z - Xn   (Z lives in W)
    for (int t = 0; t < 8; ++t) {
      int e = lane + 32 * t; int i = e >> 4, j = e & 15; int o = i * PIT + j;
      float nqr = Wr[o] - Xr[o], nqi = Wi[o] - Xi[o];
      float dr = nqr - Qr[o], di = nqi - Qi[o];
      dsum += dr * dr + di * di;
      Qr[o] = nqr; Qi[o] = nqi;
    }
    __syncthreads();
    // eps reduction (wave32) and uniform early exit
#pragma unroll
    for (int off = 16; off > 0; off >>= 1) dsum += __shfl_xor(dsum, off, 32);
    if (dsum <= TOLF) break;
  }

  // ================= finalize =================
  // amb = partial trace of X (4x4)
  if (lane < 16) {
    int j = lane >> 2, k = lane & 3;
    float sr = 0.f, si = 0.f;
#pragma unroll
    for (int a = 0; a < 4; ++a) {
      sr += Xr[(4 * a + j) * PIT + 4 * a + k];
      si += Xi[(4 * a + j) * PIT + 4 * a + k];
    }
    m4r[lane] = sr; m4i[lane] = si;
  }
  __syncthreads();
  // A2 = hermitize(4*amb); U = I
  if (lane < 16) {
    int j = lane >> 2, k = lane & 3;
    a4r[lane] = 2.f * (m4r[j * 4 + k] + m4r[k * 4 + j]);
    a4i[lane] = 2.f * (m4i[j * 4 + k] - m4i[k * 4 + j]);
    u4r[lane] = (j == k) ? 1.f : 0.f; u4i[lane] = 0.f;
  }
  __syncthreads();
  // tiny 4x4 complex Jacobi eigh, serial on lane 0
  if (lane == 0) {
    for (int sw = 0; sw < 8; ++sw) {
      for (int p = 0; p < 4; ++p) for (int q = p + 1; q < 4; ++q) {
        float app = a4r[p * 4 + p], aqq = a4r[q * 4 + q];
        float br_ = a4r[p * 4 + q], bi_ = a4i[p * 4 + q];
        float h = sqrtf(br_ * br_ + bi_ * bi_);
        if (h < 1e-22f) continue;
        float tau = (aqq - app) / (2.f * h);
        float tt = (tau >= 0.f ? 1.f : -1.f) /
                   (fabsf(tau) + sqrtf(1.f + tau * tau));
        float c = rsqrtf(1.f + tt * tt);
        float sg = tt * c;
        float sr = sg * br_ / h, si = sg * bi_ / h;
        for (int r = 0; r < 4; ++r) {
          float xpr = a4r[r * 4 + p], xpi = a4i[r * 4 + p];
          float xqr = a4r[r * 4 + q], xqi = a4i[r * 4 + q];
          a4r[r * 4 + p] = c * xpr + sr * xqr + si * xqi;
          a4i[r * 4 + p] = c * xpi + sr * xqi - si * xqr;
          a4r[r * 4 + q] = c * xqr - (sr * xpr - si * xpi);
          a4i[r * 4 + q] = c * xqi - (sr * xpi + si * xpr);
          float vpr = u4r[r * 4 + p], vpi = u4i[r * 4 + p];
          float vqr = u4r[r * 4 + q], vqi = u4i[r * 4 + q];
          u4r[r * 4 + p] = c * vpr + sr * vqr + si * vqi;
          u4i[r * 4 + p] = c * vpi + sr * vqi - si * vqr;
          u4r[r * 4 + q] = c * vqr - (sr * vpr - si * vpi);
          u4i[r * 4 + q] = c * vqi - (sr * vpi + si * vpr);
        }
        for (int t = 0; t < 4; ++t) {
          float ypr = a4r[p * 4 + t], ypi = a4i[p * 4 + t];
          float yqr = a4r[q * 4 + t], yqi = a4i[q * 4 + t];
          a4r[p * 4 + t] = c * ypr + sr * yqr - si * yqi;
          a4i[p * 4 + t] = c * ypi + sr * yqi + si * yqr;
          a4r[q * 4 + t] = c * yqr - (sr * ypr + si * ypi);
          a4i[q * 4 + t] = c * yqi - (sr * ypi - si * ypr);
        }
      }
    }
    for (int i2 = 0; i2 < 4; ++i2)
      w4[i2] = rsqrtf(fmaxf(a4r[i2 * 4 + i2], 1e-12f));
  }
  __syncthreads();
  // M = U * diag(1/sqrt(w)) * U^H  (4x4)
  if (lane < 16) {
    int j = lane >> 2, k = lane & 3;
    float sr = 0.f, si = 0.f;
#pragma unroll
    for (int t = 0; t < 4; ++t) {
      float ur = u4r[j * 4 + t], ui = u4i[j * 4 + t];
      float vr = u4r[k * 4 + t], vi = u4i[k * 4 + t];
      float w = w4[t];
      sr += w * (ur * vr + ui * vi);
      si += w * (ui * vr - ur * vi);
    }
    m4r[lane] = sr; m4i[lane] = si;
  }
  __syncthreads();
  // Mbig = I4 (x) M  -> W
  for (int t = 0; t < 8; ++t) {
    int e = lane + 32 * t; int i = e >> 4, j = e & 15; int o = i * PIT + j;
    float mr = 0.f, mi_ = 0.f;
    if ((i >> 2) == (j >> 2)) {
      int idx = (i & 3) * 4 + (j & 3);
      mr = m4r[idx]; mi_ = m4i[idx];
    }
    Wr[o] = mr; Wi[o] = mi_;
  }
  __syncthreads();
  // Y = Mbig * X  -> A   (WMMA)
  cmm_nn(Wr, Wi, Xr, Xi, Ar, Ai, lane);
  __syncthreads();
  // out = Y * Mbig^H  -> fragments -> global (real plane, imag plane)
  {
    v8f accr = {}; v8f acci = {};
#pragma unroll
    for (int kb = 0; kb < 4; ++kb) {
      v2f arf = frag_a(Ar, kb, lane), aif = frag_a(Ai, kb, lane);
      v2f btr = frag_bt(Wr, kb, lane), bti = frag_bt(Wi, kb, lane);
      accr = wmma4(arf, btr, accr);
      accr = wmma4(aif, bti, accr);        // Cr = Yr Mrᵀ + Yi Miᵀ
      acci = wmma4(aif, btr, acci);
      v2f narf; narf.x = -arf.x; narf.y = -arf.y;
      acci = wmma4(narf, bti, acci);       // Ci = Yi Mrᵀ - Yr Miᵀ
    }
    float* outb = out + b * 512;
    int n = lane & 15, mb = (lane >> 4) * 8;
#pragma unroll
    for (int r = 0; r < 8; ++r) {
      outb[(mb + r) * 16 + n] = accr[r];
      outb[256 + (mb + r) * 16 + n] = acci[r];
    }
  }
}

extern "C" void kernel_launch(void* const* d_in, const int* in_sizes, int n_in,
                              void* d_out, int out_size, void* d_ws, size_t ws_size,
                              hipStream_t stream) {
  const float* xre = (const float*)d_in[0];
  const float* xim = (const float*)d_in[1];
  const int* mi = (const int*)d_in[2];
  float* out = (float*)d_out;
  int nblocks = in_sizes[0] / 256;   // B matrices, one wave32 block each
  choi_dykstra_kernel<<<nblocks, 32, 0, stream>>>(xre, xim, mi, out);
}